// FragmentsizeDistribution2_64802466562900
// MI455X (gfx1250) — compile-verified
//
#include <hip/hip_runtime.h>
#include <hip/hip_bf16.h>
#include <math.h>

// ---------------------------------------------------------------------------
// FragmentsizeDistribution2: memory-bound stream + 1024-entry LUT.
//
//   out[i] = inside ? logsigmoid(L) + LUT[fs]
//                   : logsigmoid(-L) - log(TOTAL_WIDTH - WIDTH)
//   fs = |c[i,1] - c[i,0]|, inside = fs < 1024
//
// LUT[f] = sum of 4 hierarchical log-softmax gathers, precomputed once by a
// single-block kernel into d_ws. Roofline: ~100 MB HBM traffic -> ~4.3 us at
// 23.3 TB/s; purely streaming, no matmul structure, so no WMMA. CDNA5 paths
// used: async-to-LDS (ASYNCcnt) for the LUT broadcast, ds_load gathers,
// b128 coalesced streams with NT temporal hints, wave32 blocks.
// ---------------------------------------------------------------------------

#define WIDTH_BINS 1024
#define LOG_OUTSIDE_RANGE 98976.0f  // TOTAL_WIDTH - WIDTH = 100000 - 1024

// Native clang vector types (required by __builtin_nontemporal_load/store).
typedef __attribute__((ext_vector_type(4))) int   v4i;
typedef __attribute__((ext_vector_type(2))) float v2f;

__device__ __forceinline__ float lsm_row(const float* __restrict__ row, int nl, int c) {
    float m = row[0];
    for (int j = 1; j < nl; ++j) m = fmaxf(m, row[j]);
    float s = 0.f;
    for (int j = 0; j < nl; ++j) s += expf(row[j] - m);
    return row[c] - m - logf(s);
}

// One block of 1024 threads: thread f computes LUT[f]; thread 0 also writes
// the two scalar log-probs at ws[1024], ws[1025].
__global__ __launch_bounds__(1024) void build_table_kernel(
    const float* __restrict__ h0, const float* __restrict__ h1,
    const float* __restrict__ h2, const float* __restrict__ h3,
    const float* __restrict__ logit, float* __restrict__ ws) {
    const int f  = threadIdx.x;          // 0..1023
    const int i1 = f >> 4;               // level-1 flat bin (binwidth 16)
    const int i2 = f >> 1;               // level-2 flat bin (binwidth 2)
    float acc = lsm_row(h0,               8, f >> 7)        // level 0: parent 0
              + lsm_row(h1 + 8 * (i1 >> 3), 8, i1 & 7)      // level 1: (8,8)
              + lsm_row(h2 + 8 * (i2 >> 3), 8, i2 & 7)      // level 2: (64,8)
              + lsm_row(h3 + 2 * (f  >> 1), 2, f  & 1);     // level 3: (512,2)
    ws[f] = acc;                          // final binwidth == 1 -> -log(1) == 0
    if (f == 0) {
        const float li  = logit[0];
        const float l1p = log1pf(expf(-fabsf(li)));          // stable log1p(exp(-|x|))
        ws[1024] = fminf(li, 0.f)  - l1p;                            // log(sigmoid(li))
        ws[1025] = fminf(-li, 0.f) - l1p - logf(LOG_OUTSIDE_RANGE);  // log1p(-sig) - log(98976)
    }
}

// Streaming kernel: each 256-thread block handles a 2048-fragment tile.
// Per thread: 4x v4i (b128, NT) loads = 8 fragments, LDS LUT gathers,
// v2f NT stores.
__global__ __launch_bounds__(256) void frag_stream_kernel(
    const int* __restrict__ coords,      // (N,2) int32
    const float* __restrict__ tbl,       // ws: 1024 LUT + 2 scalars
    float* __restrict__ out, long long n) {
    __shared__ float s_table[WIDTH_BINS];
    const int tid = threadIdx.x;

    // --- async copy of the 4KB LUT into LDS (CDNA5 async-to-LDS path) ------
    // Low 32 bits of a generic pointer to LDS are the wave-relative LDS byte
    // offset (flat LDS aperture = {aperture_hi, lds_offset}).
    unsigned lds_off = (unsigned)(unsigned long long)(void*)s_table
                     + (unsigned)(tid * 16);
    const float* gsrc = tbl + tid * 4;   // 256 threads x 16B = 4096B
    asm volatile("global_load_async_to_lds_b128 %0, %1, off"
                 :: "v"(lds_off), "v"(gsrc)
                 : "memory");

    const float lp_in  = tbl[1024];      // reused by every block: keep RT-cached
    const float lp_out = tbl[1025];

#if __has_builtin(__builtin_amdgcn_s_wait_asynccnt)
    __builtin_amdgcn_s_wait_asynccnt(0);
#else
    asm volatile("s_wait_asynccnt 0x0" ::: "memory");
#endif
    __syncthreads();

    const long long base = (long long)blockIdx.x * 2048;   // fragment index of tile
    const v4i* cp = (const v4i*)coords;                    // one v4i = 2 fragments
    v2f*       op = (v2f*)out;

    if (base + 2048 <= n) {
        // ---- interior fast path: no bounds checks, int32 index math -------
        const int idx0 = (int)(blockIdx.x * 1024) + tid;   // v4i/v2f index
#pragma unroll
        for (int k = 0; k < 4; ++k) {
            const int  idx = idx0 + k * 256;
            const v4i  c   = __builtin_nontemporal_load(cp + idx);   // th:NT
            const int  d0  = abs(c.y - c.x);
            const int  d1  = abs(c.w - c.z);
            v2f r;
            r.x = (d0 < WIDTH_BINS) ? lp_in + s_table[d0 & 1023] : lp_out;
            r.y = (d1 < WIDTH_BINS) ? lp_in + s_table[d1 & 1023] : lp_out;
            __builtin_nontemporal_store(r, op + idx);                // th:NT
        }
    } else {
        // ---- tail path: guarded ------------------------------------------
#pragma unroll
        for (int k = 0; k < 4; ++k) {
            const long long pair = base + 2 * ((long long)tid + (long long)k * 256);
            if (pair + 1 < n) {
                const v4i c  = __builtin_nontemporal_load(cp + (pair >> 1));
                const int d0 = abs(c.y - c.x);
                const int d1 = abs(c.w - c.z);
                v2f r;
                r.x = (d0 < WIDTH_BINS) ? lp_in + s_table[d0 & 1023] : lp_out;
                r.y = (d1 < WIDTH_BINS) ? lp_in + s_table[d1 & 1023] : lp_out;
                __builtin_nontemporal_store(r, op + (pair >> 1));
            } else if (pair < n) {       // odd tail element
                const int d = abs(coords[2 * pair + 1] - coords[2 * pair]);
                out[pair] = (d < WIDTH_BINS) ? lp_in + s_table[d & 1023] : lp_out;
            }
        }
    }
}

extern "C" void kernel_launch(void* const* d_in, const int* in_sizes, int n_in,
                              void* d_out, int out_size, void* d_ws, size_t ws_size,
                              hipStream_t stream) {
    const int*   coords = (const int*)d_in[0];     // (N,2) int32
    const float* h0     = (const float*)d_in[1];   // (1,8)
    const float* h1     = (const float*)d_in[2];   // (8,8)
    const float* h2     = (const float*)d_in[3];   // (64,8)
    const float* h3     = (const float*)d_in[4];   // (512,2)
    const float* logit  = (const float*)d_in[5];   // scalar
    float*       out    = (float*)d_out;
    float*       ws     = (float*)d_ws;            // 1026 floats used

    const long long n = (long long)in_sizes[0] / 2;   // fragment count
    if (n <= 0) return;

    build_table_kernel<<<1, 1024, 0, stream>>>(h0, h1, h2, h3, logit, ws);

    const long long tiles = (n + 2047) / 2048;
    frag_stream_kernel<<<(unsigned)tiles, 256, 0, stream>>>(coords, ws, out, n);
}